// YOLOBuilder_73375221285235
// MI455X (gfx1250) — compile-verified
//
#include <hip/hip_runtime.h>
#include <hip/hip_bf16.h>

typedef __attribute__((ext_vector_type(2))) float v2f;
typedef __attribute__((ext_vector_type(8))) float v8f;

#define BATCH 16
#define NDET 25200
#define NPAD 32768        // next pow2 >= NDET
#define NCLS 80
#define TOPK 1024
#define MAXDET 300
#define CONF_THRES 0.25f
#define IOU_THRES 0.45f
#define MAX_WH 4096.0f
// scale_coords constants: gain = min(640/720, 640/1280) = 0.5
// x' = 2*x + 0  clip [0,1280] ;  y' = 2*y - 280  clip [0,720]
#define SCL 2.0f
#define OFFY -280.0f
#define CLIPX 1280.0f
#define CLIPY 720.0f

// ---------------------------------------------------------------------------
// Pass 1: one wave32 per detection row. Class argmax via shfl-xor butterfly
// (tie-break: smaller class index, matching jnp.argmax). Emits sortable u64
// key: ascending u64 order == descending score, ascending row index.
// ---------------------------------------------------------------------------
__global__ void prep_kernel(const float* __restrict__ pred,
                            unsigned long long* __restrict__ keys,
                            float4* __restrict__ boxes,
                            float* __restrict__ clsArr) {
  int wid  = (blockIdx.x * blockDim.x + threadIdx.x) >> 5;
  int lane = threadIdx.x & 31;
  int b = wid >> 15;            // NPAD == 2^15
  int n = wid & (NPAD - 1);
  if (b >= BATCH) return;
  if (n >= NDET) {              // padding entries sort to the very end
    if (lane == 0) keys[((size_t)b << 15) + n] = ~0ull;
    return;
  }
  const float* row = pred + ((size_t)(b * NDET + n)) * 85;
  __builtin_prefetch(row + 85, 0, 0);   // gfx1250 global_prefetch_b8

  float best = -1e30f; int bidx = 0;
  #pragma unroll
  for (int j = lane; j < NCLS; j += 32) {   // ascending j -> strict '>' keeps
    float v = row[5 + j];                   // the smaller index on ties
    if (v > best) { best = v; bidx = j; }
  }
  #pragma unroll
  for (int off = 16; off > 0; off >>= 1) {
    float ov = __shfl_xor(best, off, 32);
    int   oi = __shfl_xor(bidx, off, 32);
    if (ov > best || (ov == best && oi < bidx)) { best = ov; bidx = oi; }
  }
  if (lane == 0) {
    float obj  = row[4];
    float conf = obj * best;
    bool valid = (obj > CONF_THRES) && (conf > CONF_THRES);
    float sc   = valid ? conf : -1.0f;
    unsigned u = __float_as_uint(sc);
    unsigned s = (u & 0x80000000u) ? ~u : (u | 0x80000000u); // ascending-float key
    unsigned d = ~s;                                         // descending
    keys[((size_t)b << 15) + n] =
        ((unsigned long long)d << 32) | (unsigned)n;
    float x = row[0], y = row[1], w2 = row[2] * 0.5f, h2 = row[3] * 0.5f;
    boxes[b * NDET + n] = make_float4(x - w2, y - h2, x + w2, y + h2);
    clsArr[b * NDET + n] = (float)bidx;
  }
}

// ---------------------------------------------------------------------------
// Pass 2a: global-memory bitonic stage (only needed while j >= 2048).
// ---------------------------------------------------------------------------
__global__ void bitonic_stage(unsigned long long* __restrict__ keys,
                              int j, int k) {
  int tid = blockIdx.x * blockDim.x + threadIdx.x;
  int b = tid >> 15;
  int i = tid & (NPAD - 1);
  if (b >= BATCH) return;
  int l = i ^ j;
  if (l > i) {
    unsigned long long* kb = keys + ((size_t)b << 15);
    unsigned long long a = kb[i], c = kb[l];
    bool asc = ((i & k) == 0);
    if ((a > c) == asc) { kb[i] = c; kb[l] = a; }
  }
}

// ---------------------------------------------------------------------------
// Pass 2b: fused bitonic stages j = 1024 .. 1 for a given k, done entirely in
// LDS on 2048-element segments (16 KB LDS, 1024 threads per block). Replaces
// 11 global launches per k-phase with one; compare traffic goes L2 -> LDS.
// ---------------------------------------------------------------------------
__global__ void __launch_bounds__(1024)
bitonic_fused(unsigned long long* __restrict__ keys, int k) {
  __shared__ unsigned long long lk[2048];
  int seg = blockIdx.x;                       // BATCH * (NPAD/2048) segments
  int b   = seg / (NPAD / 2048);
  int s0  = (seg % (NPAD / 2048)) * 2048;
  unsigned long long* kb = keys + ((size_t)b << 15);
  int tid = threadIdx.x;
  lk[tid]        = kb[s0 + tid];
  lk[tid + 1024] = kb[s0 + tid + 1024];
  __syncthreads();
  #pragma unroll
  for (int j = 1024; j > 0; j >>= 1) {
    int i = ((tid & ~(j - 1)) << 1) | (tid & (j - 1));  // (i & j) == 0
    int l = i | j;
    bool asc = (((s0 + i) & k) == 0);        // direction from batch-global idx
    unsigned long long a = lk[i], c = lk[l];
    if ((a > c) == asc) { lk[i] = c; lk[l] = a; }
    __syncthreads();
  }
  kb[s0 + tid]        = lk[tid];
  kb[s0 + tid + 1024] = lk[tid + 1024];
}

// ---------------------------------------------------------------------------
// Pass 3: gather top-1024 per batch; decode score from key; class-offset boxes.
// ---------------------------------------------------------------------------
__global__ void gather_kernel(const unsigned long long* __restrict__ keys,
                              const float4* __restrict__ boxes,
                              const float* __restrict__ clsArr,
                              float4* __restrict__ topb,
                              float4* __restrict__ topoff,
                              float* __restrict__ topsc,
                              float* __restrict__ topcls) {
  int t = blockIdx.x * blockDim.x + threadIdx.x;
  if (t >= BATCH * TOPK) return;
  int b = t >> 10, r = t & (TOPK - 1);
  unsigned long long key = keys[((size_t)b << 15) + r];
  unsigned d = (unsigned)(key >> 32);
  unsigned s = ~d;
  unsigned u = (s & 0x80000000u) ? (s ^ 0x80000000u) : ~s;
  float sc = __uint_as_float(u);
  int idx = (int)(key & 0xFFFFFFFFu);        // always < NDET (NDET >= TOPK)
  float4 bx = boxes[b * NDET + idx];
  float  c  = clsArr[b * NDET + idx];
  float  of = c * MAX_WH;
  topb[t]   = bx;
  topoff[t] = make_float4(bx.x + of, bx.y + of, bx.z + of, bx.w + of);
  topsc[t]  = sc;
  topcls[t] = c;
}

// ---------------------------------------------------------------------------
// Pass 4: sequential greedy NMS, one 1024-thread block per batch, LDS state.
// Exact replica of the reference fori_loop (argmax w/ first-index tie-break,
// IoU suppression, unconditional s[p] = -1). All LDS score writes own-slot.
// ---------------------------------------------------------------------------
__global__ void __launch_bounds__(1024)
nms_kernel(const float4* __restrict__ topoff,
           const float* __restrict__ topsc,
           int* __restrict__ sel, int* __restrict__ selv) {
  __shared__ float  s[TOPK];
  __shared__ float4 offb[TOPK];
  __shared__ float  rv[TOPK];
  __shared__ int    ri[TOPK];
  int b = blockIdx.x, tid = threadIdx.x;
  s[tid]    = topsc[b * TOPK + tid];
  offb[tid] = topoff[b * TOPK + tid];
  __syncthreads();
  for (int it = 0; it < MAXDET; ++it) {
    rv[tid] = s[tid]; ri[tid] = tid;
    __syncthreads();
    for (int str = 512; str > 0; str >>= 1) {
      if (tid < str) {
        if (rv[tid + str] > rv[tid] ||
            (rv[tid + str] == rv[tid] && ri[tid + str] < ri[tid])) {
          rv[tid] = rv[tid + str]; ri[tid] = ri[tid + str];
        }
      }
      __syncthreads();
    }
    int p = ri[0]; float pv = rv[0];
    if (tid == 0) {
      sel[b * MAXDET + it]  = p;
      selv[b * MAXDET + it] = (pv > 0.0f) ? 1 : 0;
    }
    float4 P = offb[p];
    float4 Q = offb[tid];
    float ltx = fmaxf(P.x, Q.x), lty = fmaxf(P.y, Q.y);
    float rbx = fminf(P.z, Q.z), rby = fminf(P.w, Q.w);
    float w = fmaxf(rbx - ltx, 0.0f), h = fmaxf(rby - lty, 0.0f);
    float inter = w * h;
    float a1 = (P.z - P.x) * (P.w - P.y);
    float a2 = (Q.z - Q.x) * (Q.w - Q.y);
    float iou = inter / (a1 + a2 - inter + 1e-9f);
    if (iou > IOU_THRES) s[tid] = -1.0f;   // own-slot write only
    if (tid == p)        s[tid] = -1.0f;   // own-slot write only
    __syncthreads();
  }
}

// ---------------------------------------------------------------------------
// Pass 5: scale_coords via V_WMMA_F32_16X16X4_F32.
//   D = A(16x4 boxes) x B(4x16, diag 2.0) + C(broadcast (0,-280,0,-280))
// One wave per 16-row tile; lanes 0-3 / 16-19 clip+write coords, lanes
// 4-5 / 20-21 write score/class; invalid rows -> zeros (all 6 fields).
// ---------------------------------------------------------------------------
__global__ void __launch_bounds__(32)
out_wmma_kernel(const float4* __restrict__ topb,
                const float* __restrict__ topsc,
                const float* __restrict__ topcls,
                const int* __restrict__ sel,
                const int* __restrict__ selv,
                float* __restrict__ out) {
  int b    = blockIdx.x / 19;
  int tile = blockIdx.x % 19;
  int lane = threadIdx.x;        // 0..31, full wave active (EXEC all ones)
  int n    = lane & 15;
  bool hi  = lane >= 16;

  // A matrix (16x4): lanes 0-15 -> K=0,1 ; lanes 16-31 -> K=2,3 (ISA layout)
  int mrow = tile * 16 + n;
  int mc   = (mrow < MAXDET) ? mrow : (MAXDET - 1);
  int jsel = sel[b * MAXDET + mc];
  float4 box = topb[b * TOPK + jsel];
  v2f A; A[0] = hi ? box.z : box.x; A[1] = hi ? box.w : box.y;

  // B matrix (4x16): VGPR v, lanes 0-15 -> K=v ; lanes 16-31 -> K=2+v
  int k0 = hi ? 2 : 0, k1 = hi ? 3 : 1;
  v2f Bm; Bm[0] = (n == k0) ? SCL : 0.0f; Bm[1] = (n == k1) ? SCL : 0.0f;

  // C matrix: column-n offset, identical across all 8 row-VGPRs
  float cn = (n == 1 || n == 3) ? OFFY : 0.0f;
  v8f C = {cn, cn, cn, cn, cn, cn, cn, cn};

  v8f D = __builtin_amdgcn_wmma_f32_16x16x4_f32(
      false, A, false, Bm, (short)0, C, false, false);

  // coords: lane n (n<4) holds D[m][n] for m = v (+8 if hi) in VGPR v
  if (n < 4) {
    float clipHi = (n & 1) ? CLIPY : CLIPX;
    #pragma unroll
    for (int v = 0; v < 8; ++v) {
      int row = tile * 16 + (hi ? 8 + v : v);
      if (row < MAXDET) {
        float val = 0.0f;
        if (selv[b * MAXDET + row]) {
          float dv = D[v];
          val = fminf(fmaxf(dv, 0.0f), clipHi);
        }
        out[(size_t)(b * MAXDET + row) * 6 + n] = val;
      }
    }
  }
  // score (field 4) and class (field 5)
  if (n == 4 || n == 5) {
    #pragma unroll
    for (int v = 0; v < 8; ++v) {
      int row = tile * 16 + (hi ? 8 + v : v);
      if (row < MAXDET) {
        float val = 0.0f;
        if (selv[b * MAXDET + row]) {
          int j2 = sel[b * MAXDET + row];
          val = (n == 4) ? topsc[b * TOPK + j2] : topcls[b * TOPK + j2];
        }
        out[(size_t)(b * MAXDET + row) * 6 + n] = val;
      }
    }
  }
}

// ---------------------------------------------------------------------------
extern "C" void kernel_launch(void* const* d_in, const int* in_sizes, int n_in,
                              void* d_out, int out_size, void* d_ws, size_t ws_size,
                              hipStream_t stream) {
  (void)in_sizes; (void)n_in; (void)out_size; (void)ws_size;
  const float* pred = (const float*)d_in[0];
  float* out = (float*)d_out;

  // carve workspace (all offsets 16B aligned)
  char* ws = (char*)d_ws;
  size_t o = 0;
  unsigned long long* keys = (unsigned long long*)(ws + o); o += (size_t)BATCH * NPAD * 8;          // 4 MB
  float4* boxes  = (float4*)(ws + o); o += (size_t)BATCH * NDET * 16;                               // 6.3 MB
  float*  clsArr = (float*) (ws + o); o += (size_t)BATCH * NDET * 4;                                // 1.6 MB
  float4* topb   = (float4*)(ws + o); o += (size_t)BATCH * TOPK * 16;
  float4* topoff = (float4*)(ws + o); o += (size_t)BATCH * TOPK * 16;
  float*  topsc  = (float*) (ws + o); o += (size_t)BATCH * TOPK * 4;
  float*  topcls = (float*) (ws + o); o += (size_t)BATCH * TOPK * 4;
  int*    sel    = (int*)   (ws + o); o += (size_t)BATCH * MAXDET * 4;
  int*    selv   = (int*)   (ws + o); o += (size_t)BATCH * MAXDET * 4;

  // 1) decode + score + sort-key (one wave per padded row)
  {
    int waves = BATCH * NPAD;               // 524288 waves
    int blocks = waves * 32 / 256;          // 65536 blocks of 256
    prep_kernel<<<blocks, 256, 0, stream>>>(pred, keys, boxes, clsArr);
  }
  // 2) bitonic sort (ascending u64 == descending score, stable):
  //    global stages only while j >= 2048, then one fused LDS kernel per k.
  {
    int gblocks = (BATCH * NPAD) / 256;     // 2048
    int fblocks = BATCH * (NPAD / 2048);    // 256 segments
    for (int k = 2; k <= NPAD; k <<= 1) {
      for (int j = k >> 1; j >= 2048; j >>= 1)
        bitonic_stage<<<gblocks, 256, 0, stream>>>(keys, j, k);
      bitonic_fused<<<fblocks, 1024, 0, stream>>>(keys, k);
    }
  }
  // 3) gather top-1024
  gather_kernel<<<(BATCH * TOPK) / 256, 256, 0, stream>>>(
      keys, boxes, clsArr, topb, topoff, topsc, topcls);
  // 4) greedy NMS (one block per batch)
  nms_kernel<<<BATCH, 1024, 0, stream>>>(topoff, topsc, sel, selv);
  // 5) WMMA affine rescale + emit (16 batches x 19 tiles, one wave each)
  out_wmma_kernel<<<BATCH * 19, 32, 0, stream>>>(topb, topsc, topcls, sel, selv, out);
}